// TorchOLA_21466246546047
// MI455X (gfx1250) — compile-verified
//
#include <hip/hip_runtime.h>

// ---------------------------------------------------------------------------
// Overlap-add (OLA): inputs [B=32, NF=4000, FS=320] f32, shift S=160.
// Each output sample has <=2 contributions (FS == 2*S):
//   out[b,t] = (in[b,t/S,t%S]*v0 + in[b,t/S-1,t%S+S]*v1) / (v0+v1)
// Memory-bound: ~246 MB one-shot stream -> ~10.5 us roofline @ 23.3 TB/s.
// CDNA5 path: b128 async global->LDS staging (ASYNCcnt), two-tap LDS gather,
// non-temporal b128 stores. WMMA is inapplicable (zero arithmetic intensity).
// ---------------------------------------------------------------------------

#define BATCH   32
#define NFRAMES 4000
#define FSIZE   320
#define SHIFT   160
#define SIGLEN  ((NFRAMES - 1) * SHIFT + FSIZE)   // 640160
#define LBLK    32                                 // shift-blocks per chunk
#define CHUNK   (LBLK * SHIFT)                     // 5120 floats per chunk
#define NCHUNK  ((SIGLEN + CHUNK - 1) / CHUNK)     // 126 (last chunk = 160)
#define NFR_CH  (LBLK + 1)                         // 33 frames staged per chunk
#define LDSF    (NFR_CH * FSIZE)                   // 10560 floats = 42240 B
#define NB128   (LDSF / 4)                         // 2640 b128 transfers
#define THREADS 256

typedef float float4v __attribute__((ext_vector_type(4)));

__device__ __forceinline__ void async_b128(unsigned laddr, const void* gsrc) {
  asm volatile("global_load_async_to_lds_b128 %0, %1, off th:TH_LOAD_NT"
               :
               : "v"(laddr), "v"((unsigned long long)(size_t)gsrc)
               : "memory");
}

__global__ __launch_bounds__(THREADS) void ola_async_kernel(
    const float* __restrict__ in, float* __restrict__ out) {
  __shared__ float smem[LDSF];

  const int tid   = threadIdx.x;
  const int blk   = blockIdx.x;
  const int b     = blk / NCHUNK;
  const int chunk = blk % NCHUNK;

  const int  fbase = chunk * LBLK - 1;             // frame held in LDS slot 0
  const long long T0 = (long long)chunk * CHUNK;   // first output sample
  const float* inB = in + (size_t)b * NFRAMES * FSIZE;
  const unsigned lds_base = (unsigned)(size_t)(&smem[0]); // low 32b = LDS offset

  const bool interior_stage = (fbase >= 0) && (fbase + NFR_CH - 1 <= NFRAMES - 1);

  // ---- Stage NFR_CH frames into LDS via async b128 (ASYNCcnt path) ----
  if (interior_stage) {
    // Frames are contiguous in memory: straight linear copy, no div/mod.
    const char* gsrc = (const char*)(inB + (size_t)fbase * FSIZE) + tid * 16;
    unsigned    laddr = lds_base + (unsigned)tid * 16u;
    for (int q = tid; q < NB128; q += THREADS) {
      async_b128(laddr, gsrc);
      gsrc  += THREADS * 16;
      laddr += THREADS * 16u;
    }
  } else {
    // First/last chunk of a batch: clamp frame index (garbage slots unused).
    for (int q = tid; q < NB128; q += THREADS) {
      const int j = q / (FSIZE / 4);
      const int c = q % (FSIZE / 4);
      int f = fbase + j;
      f = f < 0 ? 0 : (f > NFRAMES - 1 ? NFRAMES - 1 : f);
      async_b128(lds_base + (unsigned)(j * FSIZE + c * 4) * 4u,
                 inB + (size_t)f * FSIZE + c * 4);
    }
  }
  asm volatile("s_wait_asynccnt 0" ::: "memory");
  __syncthreads();

  // ---- Two-tap gather from LDS, normalize, NT b128 store ----
  const long long remain = (long long)SIGLEN - T0;
  const int n4 = (int)((remain < (long long)CHUNK ? remain : (long long)CHUNK) / 4);
  float* outB = out + (size_t)b * SIGLEN + T0;

  // Interior chunks: both taps valid everywhere -> uniform scale 0.5.
  const bool interior_cnt = (fbase >= 0) && (fbase + LBLK <= NFRAMES - 1);

  if (interior_cnt) {
    for (int p4 = tid; p4 < n4; p4 += THREADS) {
      const int p    = p4 * 4;
      const int jblk = p / SHIFT;
      const int o    = p - jblk * SHIFT;
      const float4v a = *(const float4v*)&smem[(jblk + 1) * FSIZE + o];
      const float4v c = *(const float4v*)&smem[jblk * FSIZE + o + SHIFT];
      __builtin_nontemporal_store((a + c) * 0.5f, (float4v*)(outB + p));
    }
  } else {
    for (int p4 = tid; p4 < n4; p4 += THREADS) {
      const int p    = p4 * 4;
      const int jblk = p / SHIFT;
      const int o    = p - jblk * SHIFT;
      const int f0   = fbase + 1 + jblk;           // frame of tap A
      const bool v0  = (f0 <= NFRAMES - 1);
      const bool v1  = (f0 >= 1);
      const float4v z = {0.f, 0.f, 0.f, 0.f};
      const float4v a = v0 ? *(const float4v*)&smem[(jblk + 1) * FSIZE + o] : z;
      const float4v c = v1 ? *(const float4v*)&smem[jblk * FSIZE + o + SHIFT] : z;
      const float   s = (v0 && v1) ? 0.5f : 1.0f;
      __builtin_nontemporal_store((a + c) * s, (float4v*)(outB + p));
    }
  }
}

extern "C" void kernel_launch(void* const* d_in, const int* in_sizes, int n_in,
                              void* d_out, int out_size, void* d_ws, size_t ws_size,
                              hipStream_t stream) {
  (void)in_sizes; (void)n_in; (void)out_size; (void)d_ws; (void)ws_size;
  const float* in = (const float*)d_in[0];   // [32, 4000, 320] f32
  float* out = (float*)d_out;                // [32, 640160] f32
  const int grid = BATCH * NCHUNK;           // 4032 blocks
  ola_async_kernel<<<grid, THREADS, 0, stream>>>(in, out);
}